// GNNLayer_84963043049485
// MI455X (gfx1250) — compile-verified
//
#include <hip/hip_runtime.h>
#include <hip/hip_bf16.h>

typedef __attribute__((ext_vector_type(2))) float v2f;
typedef __attribute__((ext_vector_type(8))) float v8f;

#define NODE 512
#define NGRAPH 64
#define NTOT (NODE * NGRAPH)
#define KEDGE 16
#define DIM 64
#define LDS_STRIDE 66   // pad so gathered rows spread over LDS banks
#define NEG_SLOPE 0.2f
#define BN_EPS 1e-5f

// ---------------------------------------------------------------------------
// Kernel A: one workgroup per graph (512 nodes).
//   Phase 1: xl = x_g @ W via v_wmma_f32_16x16x4_f32, result -> LDS
//   Phase 2: per-node attention scalars a_i, a_j (a_j shared via LDS)
//   Phase 3: 17-way softmax + alpha-weighted LDS gather-aggregate, +bias,
//            write pre-BN rows to d_out
// ---------------------------------------------------------------------------
__global__ __launch_bounds__(512, 1) void gnn_graph_kernel(
    const float* __restrict__ x, const int* __restrict__ edge_src,
    const float* __restrict__ emb, const float* __restrict__ Wm,
    const float* __restrict__ att_i, const float* __restrict__ att_j,
    const float* __restrict__ att_em_i, const float* __restrict__ att_em_j,
    const float* __restrict__ bias, float* __restrict__ pre) {
  __shared__ float lds_xl[NODE * LDS_STRIDE];  // 132 KB
  __shared__ float lds_aj[NODE];

  const int g = blockIdx.x;
  const int n0 = g * NODE;
  const int tid = threadIdx.x;
  const int wave = tid >> 5;
  const int lane = tid & 31;
  const int lo = lane & 15;   // M / N index within fragment
  const int hi = lane >> 4;   // K-half select

  // ---- Phase 1: GEMM (M=512, N=64, K=64), 128 16x16 tiles over 16 waves ----
  for (int tile = wave; tile < 128; tile += 16) {
    const int mt = tile >> 2;         // 0..31
    const int nt = tile & 3;          // 0..3
    const int row = n0 + mt * 16 + lo;
    const int col = nt * 16 + lo;
    v8f acc = {};
#pragma unroll
    for (int k = 0; k < DIM; k += 4) {
      // A fragment: 16x4 fp32. lane(lo,hi) holds x[row][k+2*hi .. k+2*hi+1]
      v2f a = *(const v2f*)(x + row * DIM + k + hi * 2);
      // B fragment: 4x16 fp32. lane(lo,hi) holds W[k+2*hi+{0,1}][col]
      v2f b;
      b.x = Wm[(k + hi * 2) * DIM + col];
      b.y = Wm[(k + hi * 2 + 1) * DIM + col];
      acc = __builtin_amdgcn_wmma_f32_16x16x4_f32(false, a, false, b,
                                                  (short)0, acc, false, false);
    }
    // D layout: acc[i] = (M = i + 8*hi, N = lo)
#pragma unroll
    for (int i = 0; i < 8; ++i) {
      const int r = mt * 16 + hi * 8 + i;
      lds_xl[r * LDS_STRIDE + nt * 16 + lo] = acc[i];
    }
  }
  __syncthreads();

  // ---- Phase 2: a_i / a_j per node (one thread per node) ----
  float ai = 0.f, aj = 0.f;
  {
    const float* xr = lds_xl + tid * LDS_STRIDE;
    const float* er = emb + (n0 + tid) * DIM;
#pragma unroll 8
    for (int d = 0; d < DIM; ++d) {
      const float xv = xr[d];
      const float ev = er[d];
      ai += xv * att_i[d] + ev * att_em_i[d];
      aj += xv * att_j[d] + ev * att_em_j[d];
    }
  }
  lds_aj[tid] = aj;
  __syncthreads();

  // ---- Phase 3: softmax over 16 edges + self loop, then aggregate ----
  int src[KEDGE];
  {
    const int* ep = edge_src + (n0 + tid) * KEDGE;
#pragma unroll
    for (int e = 0; e < KEDGE; ++e) src[e] = ep[e] - n0;  // graph-local
  }
  float al[KEDGE + 1];
#pragma unroll
  for (int e = 0; e < KEDGE; ++e) {
    const float v = ai + lds_aj[src[e]];
    al[e] = v > 0.f ? v : v * NEG_SLOPE;
  }
  {
    const float v = ai + aj;  // self loop
    al[KEDGE] = v > 0.f ? v : v * NEG_SLOPE;
  }
  float mx = al[0];
#pragma unroll
  for (int e = 1; e <= KEDGE; ++e) mx = fmaxf(mx, al[e]);
  float s = 0.f;
#pragma unroll
  for (int e = 0; e <= KEDGE; ++e) {
    al[e] = __expf(al[e] - mx);
    s += al[e];
  }
  const float inv = 1.f / s;
#pragma unroll
  for (int e = 0; e <= KEDGE; ++e) al[e] *= inv;

  float* orow = pre + (n0 + tid) * DIM;
  for (int c = 0; c < 2; ++c) {  // 2 chunks of 32 features (register pressure)
    float acc[32];
#pragma unroll
    for (int i = 0; i < 32; ++i) acc[i] = 0.f;
#pragma unroll
    for (int e = 0; e <= KEDGE; ++e) {
      const int sl = (e == KEDGE) ? tid : src[e];
      const float w = al[e];
      const float* sr = lds_xl + sl * LDS_STRIDE + c * 32;
#pragma unroll
      for (int i = 0; i < 32; ++i) acc[i] += w * sr[i];
    }
#pragma unroll
    for (int i = 0; i < 32; ++i) orow[c * 32 + i] = acc[i] + bias[c * 32 + i];
  }
}

// ---------------------------------------------------------------------------
// Kernel B: per-feature batch statistics (mu, rstd) over all 32768 rows.
// One block per feature; pre-activation array is L2-resident (8 MB << 192 MB).
// ---------------------------------------------------------------------------
__global__ __launch_bounds__(256) void bn_stats_kernel(
    const float* __restrict__ pre, float* __restrict__ stats) {
  __shared__ float ssum[256];
  __shared__ float ssq[256];
  const int d = blockIdx.x;
  const int t = threadIdx.x;
  float s = 0.f, q = 0.f;
  for (int n = t; n < NTOT; n += 256) {
    const float v = pre[n * DIM + d];
    s += v;
    q += v * v;
  }
  ssum[t] = s;
  ssq[t] = q;
  __syncthreads();
  for (int o = 128; o > 0; o >>= 1) {
    if (t < o) {
      ssum[t] += ssum[t + o];
      ssq[t] += ssq[t + o];
    }
    __syncthreads();
  }
  if (t == 0) {
    const float mu = ssum[0] * (1.f / NTOT);
    const float var = ssq[0] * (1.f / NTOT) - mu * mu;
    stats[d] = mu;
    stats[DIM + d] = rsqrtf(var + BN_EPS);
  }
}

// ---------------------------------------------------------------------------
// Kernel C: in-place normalize + affine + ReLU.
// ---------------------------------------------------------------------------
__global__ __launch_bounds__(256) void bn_apply_kernel(
    float* __restrict__ out, const float* __restrict__ stats,
    const float* __restrict__ gamma, const float* __restrict__ beta) {
  const int i = blockIdx.x * 256 + threadIdx.x;
  const int d = i & (DIM - 1);
  const float v = (out[i] - stats[d]) * stats[DIM + d] * gamma[d] + beta[d];
  out[i] = fmaxf(v, 0.f);
}

extern "C" void kernel_launch(void* const* d_in, const int* in_sizes, int n_in,
                              void* d_out, int out_size, void* d_ws,
                              size_t ws_size, hipStream_t stream) {
  const float* x        = (const float*)d_in[0];
  const int*   edge     = (const int*)d_in[1];   // [2, E]; first E = src
  const float* emb      = (const float*)d_in[2];
  const float* Wm       = (const float*)d_in[3];
  const float* att_i    = (const float*)d_in[4];
  const float* att_j    = (const float*)d_in[5];
  const float* att_em_i = (const float*)d_in[6];
  const float* att_em_j = (const float*)d_in[7];
  const float* bias     = (const float*)d_in[8];
  const float* gamma    = (const float*)d_in[9];
  const float* beta     = (const float*)d_in[10];

  float* out = (float*)d_out;          // pre-BN rows, then normalized in place
  float* stats = (float*)d_ws;         // 128 floats: mu[64], rstd[64]

  gnn_graph_kernel<<<NGRAPH, NODE, 0, stream>>>(x, edge, emb, Wm, att_i, att_j,
                                                att_em_i, att_em_j, bias, out);
  bn_stats_kernel<<<DIM, 256, 0, stream>>>(out, stats);
  bn_apply_kernel<<<(NTOT * DIM) / 256, 256, 0, stream>>>(out, stats, gamma,
                                                          beta);
}